// GNNEncoderV5_69148973466096
// MI455X (gfx1250) — compile-verified
//
#include <hip/hip_runtime.h>
#include <hip/hip_bf16.h>

// ---- types for WMMA / async LDS ----
typedef __attribute__((ext_vector_type(2))) float v2f;
typedef __attribute__((ext_vector_type(8))) float v8f;
typedef __attribute__((ext_vector_type(4))) int   v4i;

#define F 64
#define H 4
#define HC 256
#define DE 16
#define NLAYERS 4
#define NEG_SLOPE 0.2f

#if __has_builtin(__builtin_amdgcn_global_load_async_to_lds_b128)
#define HAVE_ASYNC_LDS 1
#else
#define HAVE_ASYNC_LDS 0
#endif

__device__ __forceinline__ void wait_asynccnt0() {
#if __has_builtin(__builtin_amdgcn_s_wait_asynccnt)
    __builtin_amdgcn_s_wait_asynccnt(0);
#else
    asm volatile("s_wait_asynccnt 0x0" ::: "memory");
#endif
}

// float atomic max via int punning (works for full float range)
__device__ __forceinline__ void atomicMaxFloat(float* addr, float val) {
    if (val >= 0.0f) {
        atomicMax((int*)addr, __float_as_int(val));
    } else {
        atomicMin((unsigned int*)addr, (unsigned int)__float_as_int(val));
    }
}

// ---------------------------------------------------------------------------
// GEMM: Y[N,256] = X[N,64] @ W[64,256] + bias[256], fp32 WMMA 16x16x4.
// Block: 512 threads = 16 waves. Each block owns a 16-row strip of X (LDS),
// wave w computes the 16x16 output tile at columns [16w, 16w+16).
// ---------------------------------------------------------------------------
__global__ void gemm_xw_wmma(const float* __restrict__ X,
                             const float* __restrict__ W,
                             const float* __restrict__ bias,
                             float* __restrict__ Y, int N) {
    __shared__ __align__(16) float xs[16 * F];
    const int tid = threadIdx.x;
    const int row0 = blockIdx.x * 16;
    // 16x64 strip = 256 float4; one b128 per thread for tid < 256
    if (tid < 256) {
        const int r = row0 + (tid >> 4);   // 16 float4 per row
        float4 v = make_float4(0.f, 0.f, 0.f, 0.f);
        if (r < N) v = ((const float4*)(X + (size_t)row0 * F))[tid];
        ((float4*)xs)[tid] = v;
    }
    __builtin_prefetch(W + (tid & 63) * HC, 0, 0);  // pull W toward L2
    __syncthreads();

    const int wave = tid >> 5;         // 0..15 -> column tile
    const int lane = tid & 31;
    const int colb = wave * 16;
    const int m    = lane & 15;        // M index (A) / N index (B,C,D)
    const int kh   = (lane >> 4) << 1; // 0 or 2: K sub-offset for this lane half

    v8f acc = {};
#pragma unroll
    for (int s = 0; s < 16; ++s) {     // K = 64 = 16 steps of 4
        const int k0 = 4 * s + kh;
        v2f a, b;
        a.x = xs[m * F + k0];
        a.y = xs[m * F + k0 + 1];
        b.x = W[(k0    ) * HC + colb + m];
        b.y = W[(k0 + 1) * HC + colb + m];
        acc = __builtin_amdgcn_wmma_f32_16x16x4_f32(
            false, a, false, b, (short)0, acc, false, false);
    }

    const float bv = bias[colb + m];
#pragma unroll
    for (int j = 0; j < 8; ++j) {
        const int row = row0 + j + ((lane >> 4) << 3);
        if (row < N) Y[row * HC + colb + m] = acc[j] + bv;
    }
}

// ---------------------------------------------------------------------------
// edge_attr mean over E rows (for self-loop fill_value='mean')
// ---------------------------------------------------------------------------
__global__ void ea_zero_kernel(float* s) {
    if (threadIdx.x < DE) s[threadIdx.x] = 0.0f;
}

__global__ void ea_reduce_kernel(const float* __restrict__ EA, int E,
                                 float* __restrict__ s) {
    float acc[DE];
#pragma unroll
    for (int d = 0; d < DE; ++d) acc[d] = 0.0f;
    for (int e = blockIdx.x * 256 + threadIdx.x; e < E; e += gridDim.x * 256) {
        const float4* row = (const float4*)(EA + (size_t)e * DE);
#pragma unroll
        for (int k = 0; k < 4; ++k) {
            float4 v = row[k];
            acc[4 * k + 0] += v.x;
            acc[4 * k + 1] += v.y;
            acc[4 * k + 2] += v.z;
            acc[4 * k + 3] += v.w;
        }
    }
    for (int off = 16; off; off >>= 1) {
#pragma unroll
        for (int d = 0; d < DE; ++d) acc[d] += __shfl_xor(acc[d], off, 32);
    }
    __shared__ float blk[8][DE];
    const int lane = threadIdx.x & 31, wv = threadIdx.x >> 5;
    if (lane == 0) {
#pragma unroll
        for (int d = 0; d < DE; ++d) blk[wv][d] = acc[d];
    }
    __syncthreads();
    if (threadIdx.x < DE) {
        float t = 0.0f;
#pragma unroll
        for (int w = 0; w < 8; ++w) t += blk[w][threadIdx.x];
        atomicAdd(&s[threadIdx.x], t);
    }
}

__global__ void ea_scale_kernel(const float* s, float* mean, int E) {
    if (threadIdx.x < DE) mean[threadIdx.x] = s[threadIdx.x] / (float)E;
}

// ---------------------------------------------------------------------------
// Per-layer init: nmax=-FLT_MAX, nsum=0, agg=0
// ---------------------------------------------------------------------------
__global__ void init_layer_kernel(float* nmax, float* nsum, float* agg, int N) {
    const int i = blockIdx.x * 256 + threadIdx.x;
    if (i < N * H) { nmax[i] = -3.402823466e38f; nsum[i] = 0.0f; }
    if (i < N * HC) agg[i] = 0.0f;
}

// ---------------------------------------------------------------------------
// Edge scores: one wave per edge. ee = ea @ We computed on the fly (We in LDS,
// staged with async-to-LDS b128 when available). All gathers are b128.
// score[e,h] = sum_c att[h,c]*leaky_relu(xl[src]+xr[dst]+ee); atomicMax nmax.
// ---------------------------------------------------------------------------
__global__ void edge_score_kernel(const float* __restrict__ xl,
                                  const float* __restrict__ xr,
                                  const float* __restrict__ EA,
                                  const float* __restrict__ ea_mean,
                                  const float* __restrict__ We,
                                  const float* __restrict__ att,
                                  const int* __restrict__ ei,
                                  int E, int N,
                                  float* __restrict__ score,
                                  float* __restrict__ nmax) {
    __shared__ __align__(16) float ws[DE * HC];   // 16 KB
    __shared__ __align__(16) float atts[HC];      // att [H,F] -> flat 256
    const int tid = threadIdx.x;    // 256 threads = 8 waves = 8 edges
#if HAVE_ASYNC_LDS
    // 4096 floats = 1024 b128 transfers; 4 per thread, ASYNCcnt-tracked
    for (int i = tid; i < DE * HC / 4; i += 256) {
        __builtin_amdgcn_global_load_async_to_lds_b128(
            (__attribute__((address_space(1))) v4i*)(We + 4 * i),
            (__attribute__((address_space(3))) v4i*)(ws + 4 * i), 0, 0);
    }
    atts[tid] = att[tid];
    wait_asynccnt0();
#else
    for (int i = tid; i < DE * HC / 4; i += 256)
        ((float4*)ws)[i] = ((const float4*)We)[i];
    atts[tid] = att[tid];
#endif
    __syncthreads();

    const int ET = E + N;
    const int e = blockIdx.x * 8 + (tid >> 5);
    if (e >= ET) return;
    const int lane = tid & 31;

    int src, dst;
    const float* ea;
    if (e < E) { src = ei[e]; dst = ei[E + e]; ea = EA + (size_t)e * DE; }
    else       { src = dst = e - E;            ea = ea_mean; }

    float eav[DE];
#pragma unroll
    for (int k = 0; k < 4; ++k)
        *(float4*)&eav[4 * k] = ((const float4*)ea)[k];

    const int cb = lane * 8;        // 8 channels/lane; 8 lanes/head
    float xv[8];
    {
        const float4* l4 = (const float4*)(xl + (size_t)src * HC + cb);
        const float4* r4 = (const float4*)(xr + (size_t)dst * HC + cb);
        float4 a0 = l4[0], a1 = l4[1], b0 = r4[0], b1 = r4[1];
        xv[0] = a0.x + b0.x; xv[1] = a0.y + b0.y;
        xv[2] = a0.z + b0.z; xv[3] = a0.w + b0.w;
        xv[4] = a1.x + b1.x; xv[5] = a1.y + b1.y;
        xv[6] = a1.z + b1.z; xv[7] = a1.w + b1.w;
    }

    float partial = 0.0f;
#pragma unroll
    for (int q = 0; q < 8; ++q) {
        const int c = cb + q;
        float ee = 0.0f;
#pragma unroll
        for (int d = 0; d < DE; ++d) ee += eav[d] * ws[d * HC + c];
        float v = xv[q] + ee;
        v = (v > 0.0f) ? v : NEG_SLOPE * v;
        partial += v * atts[c];
    }
    // reduce over the 8 lanes of each head
    partial += __shfl_xor(partial, 1, 32);
    partial += __shfl_xor(partial, 2, 32);
    partial += __shfl_xor(partial, 4, 32);
    if ((lane & 7) == 0) {
        const int h = lane >> 3;
        score[(size_t)e * H + h] = partial;
        atomicMaxFloat(&nmax[dst * H + h], partial);
    }
}

// ---------------------------------------------------------------------------
// exp(score - max) and segment sum
// ---------------------------------------------------------------------------
__global__ void edge_exp_kernel(const int* __restrict__ ei, int E, int N,
                                const float* __restrict__ nmax,
                                float* __restrict__ score,
                                float* __restrict__ nsum) {
    const int ET = E + N;
    const int idx = blockIdx.x * 256 + threadIdx.x;
    if (idx >= ET * H) return;
    const int e = idx >> 2, h = idx & 3;
    const int dst = (e < E) ? ei[E + e] : (e - E);
    const float v = __expf(score[idx] - nmax[dst * H + h]);
    score[idx] = v;
    atomicAdd(&nsum[dst * H + h], v);
}

// ---------------------------------------------------------------------------
// agg[dst] += xl[src] * alpha  (one wave per edge, b128 gathers)
// ---------------------------------------------------------------------------
__global__ void edge_agg_kernel(const int* __restrict__ ei, int E, int N,
                                const float* __restrict__ xl,
                                const float* __restrict__ score,
                                const float* __restrict__ nsum,
                                float* __restrict__ agg) {
    const int ET = E + N;
    const int e = blockIdx.x * 8 + (threadIdx.x >> 5);
    if (e >= ET) return;
    const int lane = threadIdx.x & 31;
    int src, dst;
    if (e < E) { src = ei[e]; dst = ei[E + e]; }
    else       { src = dst = e - E; }
    const int h = lane >> 3;
    const float alpha =
        score[(size_t)e * H + h] / (nsum[dst * H + h] + 1e-16f);
    const int cb = lane * 8;
    float av[8];
    {
        const float4* l4 = (const float4*)(xl + (size_t)src * HC + cb);
        *(float4*)&av[0] = l4[0];
        *(float4*)&av[4] = l4[1];
    }
#pragma unroll
    for (int q = 0; q < 8; ++q) {
        atomicAdd(&agg[(size_t)dst * HC + cb + q], av[q] * alpha);
    }
}

// ---------------------------------------------------------------------------
// head mean + bias, relu, layernorm, residual. One wave per node (2 ch/lane).
// ---------------------------------------------------------------------------
__global__ void node_fin_kernel(const float* __restrict__ agg,
                                const float* __restrict__ bo,
                                const float* __restrict__ g,
                                const float* __restrict__ bln,
                                const float* __restrict__ hprev,
                                float* __restrict__ hnext,
                                int N, int layer) {
    const int n = blockIdx.x * 8 + (threadIdx.x >> 5);
    if (n >= N) return;
    const int lane = threadIdx.x & 31;
    const int c0 = lane, c1 = lane + 32;
    float y0 = 0.0f, y1 = 0.0f;
#pragma unroll
    for (int hh = 0; hh < H; ++hh) {
        y0 += agg[(size_t)n * HC + hh * F + c0];
        y1 += agg[(size_t)n * HC + hh * F + c1];
    }
    y0 = 0.25f * y0 + bo[c0];
    y1 = 0.25f * y1 + bo[c1];
    y0 = fmaxf(y0, 0.0f);
    y1 = fmaxf(y1, 0.0f);
    float s = y0 + y1;
    for (int off = 16; off; off >>= 1) s += __shfl_xor(s, off, 32);
    const float mu = s * (1.0f / F);
    const float d0 = y0 - mu, d1 = y1 - mu;
    float v = d0 * d0 + d1 * d1;
    for (int off = 16; off; off >>= 1) v += __shfl_xor(v, off, 32);
    const float inv = rsqrtf(v * (1.0f / F) + 1e-5f);
    float o0 = d0 * inv * g[c0] + bln[c0];
    float o1 = d1 * inv * g[c1] + bln[c1];
    if (layer > 0) {
        o0 += hprev[(size_t)n * F + c0];
        o1 += hprev[(size_t)n * F + c1];
    }
    hnext[(size_t)n * F + c0] = o0;
    hnext[(size_t)n * F + c1] = o1;
}

// ---------------------------------------------------------------------------
// final output: h followed by batch (cast)
// ---------------------------------------------------------------------------
__global__ void write_out_kernel(const float* __restrict__ h,
                                 const int* __restrict__ batch,
                                 float* __restrict__ out, int N) {
    const int i = blockIdx.x * 256 + threadIdx.x;
    if (i < N * F) out[i] = h[i];
    if (i < N) out[N * F + i] = (float)batch[i];
}

// ---------------------------------------------------------------------------
extern "C" void kernel_launch(void* const* d_in, const int* in_sizes, int n_in,
                              void* d_out, int out_size, void* d_ws, size_t ws_size,
                              hipStream_t stream) {
    const float* x    = (const float*)d_in[0];
    const float* EA   = (const float*)d_in[1];
    const float* Wl   = (const float*)d_in[2];
    const float* bl   = (const float*)d_in[3];
    const float* Wr   = (const float*)d_in[4];
    const float* br   = (const float*)d_in[5];
    const float* We   = (const float*)d_in[6];
    const float* att  = (const float*)d_in[7];
    const float* bo   = (const float*)d_in[8];
    const float* lng  = (const float*)d_in[9];
    const float* lnb  = (const float*)d_in[10];
    const int*   ei   = (const int*)d_in[11];
    const int*   bat  = (const int*)d_in[12];

    const int N  = in_sizes[0] / F;      // 20000
    const int E  = in_sizes[11] / 2;     // 160000
    const int ET = E + N;                // with self loops

    // workspace carve-up
    size_t off = 0;
    auto carve = [&](size_t bytes) -> void* {
        void* p = (char*)d_ws + off;
        off += (bytes + 255) & ~(size_t)255;
        return p;
    };
    float* xl      = (float*)carve((size_t)N * HC * 4);
    float* xr      = (float*)carve((size_t)N * HC * 4);
    float* agg     = (float*)carve((size_t)N * HC * 4);
    float* score   = (float*)carve((size_t)ET * H * 4);
    float* nmax    = (float*)carve((size_t)N * H * 4);
    float* nsum    = (float*)carve((size_t)N * H * 4);
    float* hA      = (float*)carve((size_t)N * F * 4);
    float* hB      = (float*)carve((size_t)N * F * 4);
    float* ea_sum  = (float*)carve(DE * 4);
    float* ea_mean = (float*)carve(DE * 4);

    // edge_attr mean (once)
    ea_zero_kernel<<<1, 32, 0, stream>>>(ea_sum);
    ea_reduce_kernel<<<240, 256, 0, stream>>>(EA, E, ea_sum);
    ea_scale_kernel<<<1, 32, 0, stream>>>(ea_sum, ea_mean, E);

    // h0 = x
    (void)hipMemcpyAsync(hA, x, (size_t)N * F * 4, hipMemcpyDeviceToDevice, stream);

    float* hcur = hA;
    float* hnxt = hB;

    const int gemm_blocks = (N + 15) / 16;
    const int edge_blocks = (ET + 7) / 8;
    const int exp_blocks  = (ET * H + 255) / 256;
    const int init_blocks = (N * HC + 255) / 256;
    const int node_blocks = (N + 7) / 8;

    for (int i = 0; i < NLAYERS; ++i) {
        const float* Wl_i  = Wl  + (size_t)i * F * HC;
        const float* bl_i  = bl  + (size_t)i * HC;
        const float* Wr_i  = Wr  + (size_t)i * F * HC;
        const float* br_i  = br  + (size_t)i * HC;
        const float* We_i  = We  + (size_t)i * DE * HC;
        const float* att_i = att + (size_t)i * H * F;
        const float* bo_i  = bo  + (size_t)i * F;
        const float* g_i   = lng + (size_t)i * F;
        const float* b_i   = lnb + (size_t)i * F;

        gemm_xw_wmma<<<gemm_blocks, 512, 0, stream>>>(hcur, Wl_i, bl_i, xl, N);
        gemm_xw_wmma<<<gemm_blocks, 512, 0, stream>>>(hcur, Wr_i, br_i, xr, N);
        init_layer_kernel<<<init_blocks, 256, 0, stream>>>(nmax, nsum, agg, N);
        edge_score_kernel<<<edge_blocks, 256, 0, stream>>>(
            xl, xr, EA, ea_mean, We_i, att_i, ei, E, N, score, nmax);
        edge_exp_kernel<<<exp_blocks, 256, 0, stream>>>(ei, E, N, nmax, score, nsum);
        edge_agg_kernel<<<edge_blocks, 256, 0, stream>>>(
            ei, E, N, xl, score, nsum, agg);
        node_fin_kernel<<<node_blocks, 256, 0, stream>>>(
            agg, bo_i, g_i, b_i, hcur, hnxt, N, i);

        float* t = hcur; hcur = hnxt; hnxt = t;
    }

    const int out_blocks = (N * F + 255) / 256;
    write_out_kernel<<<out_blocks, 256, 0, stream>>>(hcur, bat, (float*)d_out, N);
}